// Cross_Attention_55224689492544
// MI455X (gfx1250) — compile-verified
//
#include <hip/hip_runtime.h>

// ---------------------------------------------------------------------------
// Problem constants (from reference: B=16, C=256, H=128, W=128, fp32)
// ---------------------------------------------------------------------------
constexpr int kB = 16, kC = 256, kH = 128, kW = 128;
constexpr float kEps = 1e-5f;

constexpr size_t BHW = (size_t)kB * kH * kW;        // 262144
constexpr size_t BCH = (size_t)kB * kC * kH;        // 524288
constexpr size_t BCW = (size_t)kB * kC * kW;        // 524288
constexpr size_t ATTN_PER = (size_t)kW * kC;        // 32768 (max sim size)
constexpr size_t ATT = (size_t)kB * 3 * ATTN_PER;

// workspace layout (in floats) -- ~27.3 MB total, L2-resident on MI455X
constexpr size_t O_XHW = 0;
constexpr size_t O_XCH = O_XHW + BHW;
constexpr size_t O_XCW = O_XCH + BCH;
constexpr size_t O_ATT = O_XCW + BCW;
constexpr size_t O_Y12 = O_ATT + ATT;
constexpr size_t O_Y13 = O_Y12 + BHW;
constexpr size_t O_Y21 = O_Y13 + BHW;
constexpr size_t O_Y23 = O_Y21 + BCH;
constexpr size_t O_Y31 = O_Y23 + BCH;
constexpr size_t O_Y32 = O_Y31 + BCW;
constexpr size_t O_ST  = O_Y32 + BCW;     // [0..3]: mean/rstd y12,y13; [4 + (t*C+c)*2]
constexpr size_t O_GA  = O_ST + 4096;     // g12*g13  [B,H,W]
constexpr size_t O_GB  = O_GA + BHW;      // g21*g23  [B,C,H]
constexpr size_t O_GC  = O_GB + BCH;      // g31*g32  [B,C,W]

typedef float v2f __attribute__((ext_vector_type(2)));
typedef float v8f __attribute__((ext_vector_type(8)));

__device__ __forceinline__ float sigmoidf_(float x) { return 1.0f / (1.0f + __expf(-x)); }

// ---------------------------------------------------------------------------
// Pass A: x_hw[b,h,w] = max_c x[b,c,h,w]
// One thread owns 4 consecutive w -> global_load_b128 per channel iteration,
// fully coalesced streaming of the 256 MB tensor at DRAM rate.
// ---------------------------------------------------------------------------
__global__ void k_pool_hw(const float* __restrict__ x, float* __restrict__ xhw) {
  size_t i4 = (size_t)blockIdx.x * blockDim.x + threadIdx.x;
  size_t base = i4 * 4;                          // index into [B,H,W]
  if (base >= BHW) return;
  size_t b = base / ((size_t)kH * kW);
  size_t hw = base % ((size_t)kH * kW);
  const float* p = x + b * kC * kH * kW + hw;
  float4 m = make_float4(-3.402823466e38f, -3.402823466e38f,
                         -3.402823466e38f, -3.402823466e38f);
#pragma unroll 4
  for (int c = 0; c < kC; ++c) {
    float4 v = *(const float4*)(p + (size_t)c * kH * kW);
    m.x = fmaxf(m.x, v.x);
    m.y = fmaxf(m.y, v.y);
    m.z = fmaxf(m.z, v.z);
    m.w = fmaxf(m.w, v.w);
  }
  *(float4*)(xhw + base) = m;
}

// ---------------------------------------------------------------------------
// Pass B: per (b,c) plane -> x_ch[b,c,h] (row max), x_cw[b,c,w] (col max).
// Second scan hits L0/L2 (plane = 64 KB).
// ---------------------------------------------------------------------------
__global__ void k_pool_chw(const float* __restrict__ x, float* __restrict__ xch,
                           float* __restrict__ xcw) {
  int b = blockIdx.x / kC, c = blockIdx.x % kC;
  int t = threadIdx.x;  // 128 threads
  const float* plane = x + ((size_t)b * kC + c) * kH * kW;
  // row max: thread t = row h, contiguous float4 reads
  const float4* row = (const float4*)(plane + (size_t)t * kW);
  float m = -3.402823466e38f;
#pragma unroll 8
  for (int i = 0; i < kW / 4; ++i) {
    float4 v = row[i];
    m = fmaxf(m, fmaxf(fmaxf(v.x, v.y), fmaxf(v.z, v.w)));
  }
  xch[((size_t)b * kC + c) * kH + t] = m;
  // col max: thread t = column w, lane-coalesced reads (cache-resident)
  float mc = -3.402823466e38f;
#pragma unroll 8
  for (int h = 0; h < kH; ++h) mc = fmaxf(mc, plane[(size_t)h * kW + t]);
  xcw[((size_t)b * kC + c) * kW + t] = mc;
}

// ---------------------------------------------------------------------------
// Generic strided GEMM with V_WMMA_F32_16X16X4_F32  (D = A(MxK) * B(KxN)).
// A element (m,k) at A + m*sa_m + k*sa_k ; B element (k,n) at B + k*sb_k + n*sb_n.
// Fragment layouts per CDNA5 ISA 7.12.2:
//   A 16x4: lane<16 -> {K=k0,k0+1}, lane>=16 -> {K=k0+2,k0+3}, M = lane%16
//   B 4x16: mirrored (N = lane%16)
//   C/D 16x16: VGPR v -> M = v + 8*(lane/16), N = lane%16
// Wave-uniform control flow: EXEC is all-ones at every WMMA issue.
// ---------------------------------------------------------------------------
__device__ __forceinline__ void wmma_gemm(const float* __restrict__ A, int sa_m, int sa_k,
                                          const float* __restrict__ Bm, int sb_k, int sb_n,
                                          float* __restrict__ D, int ld_d,
                                          int M, int N, int K,
                                          int wid, int nwaves, int lane) {
  const int tiles_n = N >> 4;
  const int ntiles = (M >> 4) * tiles_n;
  const int row = lane & 15, half = lane >> 4;
  for (int t = wid; t < ntiles; t += nwaves) {
    const int m0 = (t / tiles_n) << 4, n0 = (t % tiles_n) << 4;
    v8f acc = {0.f, 0.f, 0.f, 0.f, 0.f, 0.f, 0.f, 0.f};
    const float* Ab = A + (size_t)(m0 + row) * sa_m;
    const float* Bb = Bm + (size_t)(n0 + row) * sb_n;
#pragma unroll 4
    for (int k0 = 0; k0 < K; k0 += 4) {
      const int ka = k0 + 2 * half;
      v2f a, b;
      a.x = Ab[(size_t)ka * sa_k];
      a.y = Ab[(size_t)(ka + 1) * sa_k];
      b.x = Bb[(size_t)ka * sb_k];
      b.y = Bb[(size_t)(ka + 1) * sb_k];
      acc = __builtin_amdgcn_wmma_f32_16x16x4_f32(false, a, false, b, (short)0, acc,
                                                  false, false);
    }
#pragma unroll
    for (int v = 0; v < 8; ++v)
      D[(size_t)(m0 + v + 8 * half) * ld_d + n0 + row] = acc[v];
  }
}

// ---------------------------------------------------------------------------
// Pass C: one block per (batch, branch): sim GEMM -> flat softmax -> 2 GEMMs.
// 512 threads = 16 wave32 waves for L2-latency hiding on fragment gathers.
// ---------------------------------------------------------------------------
__global__ void k_attn(float* __restrict__ ws) {
  const int b = blockIdx.x / 3, br = blockIdx.x % 3;
  const float* xhw = ws + O_XHW + (size_t)b * kH * kW;
  const float* xch = ws + O_XCH + (size_t)b * kC * kH;
  const float* xcw = ws + O_XCW + (size_t)b * kC * kW;
  float* attn = ws + O_ATT + ((size_t)b * 3 + br) * ATTN_PER;

  const int tid = threadIdx.x;
  const int lane = tid & 31, wid = tid >> 5, nw = blockDim.x >> 5;

  int nelem;
  if (br == 0) {  // sim12[w,c] = sum_h xhw[h,w]*xch[c,h]
    wmma_gemm(xhw, 1, kW, xch, 1, kH, attn, kC, kW, kC, kH, wid, nw, lane);
    nelem = kW * kC;
  } else if (br == 1) {  // sim13[h,c] = sum_w xhw[h,w]*xcw[c,w]
    wmma_gemm(xhw, kW, 1, xcw, 1, kW, attn, kC, kH, kC, kW, wid, nw, lane);
    nelem = kH * kC;
  } else {  // sim23[h,w] = sum_c xch[c,h]*xcw[c,w]
    wmma_gemm(xch, 1, kH, xcw, kW, 1, attn, kW, kH, kW, kC, wid, nw, lane);
    nelem = kH * kW;
  }
  __syncthreads();

  // flat softmax over all nelem entries (nelem % blockDim == 0 -> no divergence)
  __shared__ float red[512];
  float lm = -3.402823466e38f;
  for (int i = tid; i < nelem; i += blockDim.x) lm = fmaxf(lm, attn[i]);
  red[tid] = lm;
  __syncthreads();
  for (int s = 256; s > 0; s >>= 1) {
    if (tid < s) red[tid] = fmaxf(red[tid], red[tid + s]);
    __syncthreads();
  }
  const float gmax = red[0];
  __syncthreads();
  float ls = 0.f;
  for (int i = tid; i < nelem; i += blockDim.x) ls += expf(attn[i] - gmax);
  red[tid] = ls;
  __syncthreads();
  for (int s = 256; s > 0; s >>= 1) {
    if (tid < s) red[tid] += red[tid + s];
    __syncthreads();
  }
  const float inv = 1.0f / red[0];
  __syncthreads();
  for (int i = tid; i < nelem; i += blockDim.x) attn[i] = expf(attn[i] - gmax) * inv;
  __syncthreads();

  if (br == 0) {
    // y12[h,w] = sum_c attn12[w,c]*xch[c,h]
    wmma_gemm(xch, 1, kH, attn, 1, kC, ws + O_Y12 + (size_t)b * kH * kW, kW,
              kH, kW, kC, wid, nw, lane);
    // y21[c,h] = sum_w attn12[w,c]*xhw[h,w]
    wmma_gemm(attn, 1, kC, xhw, 1, kW, ws + O_Y21 + (size_t)b * kC * kH, kH,
              kC, kH, kW, wid, nw, lane);
  } else if (br == 1) {
    // y13[h,w] = sum_c attn13[h,c]*xcw[c,w]
    wmma_gemm(attn, kC, 1, xcw, kW, 1, ws + O_Y13 + (size_t)b * kH * kW, kW,
              kH, kW, kC, wid, nw, lane);
    // y31[c,w] = sum_h attn13[h,c]*xhw[h,w]
    wmma_gemm(attn, 1, kC, xhw, kW, 1, ws + O_Y31 + (size_t)b * kC * kW, kW,
              kC, kW, kH, wid, nw, lane);
  } else {
    // y23[c,h] = sum_w attn23[h,w]*xcw[c,w]
    wmma_gemm(xcw, kW, 1, attn, 1, kW, ws + O_Y23 + (size_t)b * kC * kH, kH,
              kC, kH, kW, wid, nw, lane);
    // y32[c,w] = sum_h attn23[h,w]*xch[c,h]
    wmma_gemm(xch, kH, 1, attn, kW, 1, ws + O_Y32 + (size_t)b * kC * kW, kW,
              kC, kW, kH, wid, nw, lane);
  }
}

// ---------------------------------------------------------------------------
// Pass D1: scalar BN stats for y12 (block 0) and y13 (block 1): whole-tensor
// biased mean/var over B*H*W.
// ---------------------------------------------------------------------------
__global__ void k_bn_scalar(float* __restrict__ ws) {
  const float* y = ws + (blockIdx.x == 0 ? O_Y12 : O_Y13);
  __shared__ float rs[256], rq[256];
  const int tid = threadIdx.x;
  float s = 0.f, q = 0.f;
  for (size_t i = tid; i < BHW; i += blockDim.x) {
    float v = y[i];
    s += v;
    q += v * v;
  }
  rs[tid] = s; rq[tid] = q;
  __syncthreads();
  for (int st = 128; st > 0; st >>= 1) {
    if (tid < st) { rs[tid] += rs[tid + st]; rq[tid] += rq[tid + st]; }
    __syncthreads();
  }
  if (tid == 0) {
    const float n = (float)BHW;
    float mean = rs[0] / n;
    float var = fmaxf(rq[0] / n - mean * mean, 0.f);
    ws[O_ST + 2 * blockIdx.x + 0] = mean;
    ws[O_ST + 2 * blockIdx.x + 1] = rsqrtf(var + kEps);
  }
}

// ---------------------------------------------------------------------------
// Pass D2: per-channel BN stats for y21/y23/y31/y32 (stats over B * 128).
// grid = 4*C blocks of 128 threads.
// ---------------------------------------------------------------------------
__global__ void k_bn_chan(float* __restrict__ ws) {
  constexpr size_t yoff[4] = {O_Y21, O_Y23, O_Y31, O_Y32};
  const int t = blockIdx.x / kC, c = blockIdx.x % kC;
  const float* y = ws + yoff[t];
  const int tid = threadIdx.x;  // position within H (or W) = 128
  __shared__ float rs[128], rq[128];
  float s = 0.f, q = 0.f;
#pragma unroll
  for (int b = 0; b < kB; ++b) {
    float v = y[((size_t)b * kC + c) * 128 + tid];
    s += v;
    q += v * v;
  }
  rs[tid] = s; rq[tid] = q;
  __syncthreads();
  for (int st = 64; st > 0; st >>= 1) {
    if (tid < st) { rs[tid] += rs[tid + st]; rq[tid] += rq[tid + st]; }
    __syncthreads();
  }
  if (tid == 0) {
    const float n = (float)(kB * 128);
    float mean = rs[0] / n;
    float var = fmaxf(rq[0] / n - mean * mean, 0.f);
    ws[O_ST + 4 + ((size_t)t * kC + c) * 2 + 0] = mean;
    ws[O_ST + 4 + ((size_t)t * kC + c) * 2 + 1] = rsqrtf(var + kEps);
  }
}

// ---------------------------------------------------------------------------
// Gate precompute: GA = sig(bn(y12))*sig(bn(y13))  over [B,H,W]
// ---------------------------------------------------------------------------
__global__ void k_gate_hw(float* __restrict__ ws, const float* __restrict__ w1,
                          const float* __restrict__ b1) {
  size_t i = (size_t)blockIdx.x * blockDim.x + threadIdx.x;
  if (i >= BHW) return;
  const float m12 = ws[O_ST + 0], r12 = ws[O_ST + 1];
  const float m13 = ws[O_ST + 2], r13 = ws[O_ST + 3];
  const float wv = w1[0], bv = b1[0];
  float g = sigmoidf_((ws[O_Y12 + i] - m12) * r12 * wv + bv) *
            sigmoidf_((ws[O_Y13 + i] - m13) * r13 * wv + bv);
  ws[O_GA + i] = g;
}

// GB = sig(bn(y21))*sig(bn(y23)) over [B,C,H]  (which=0)
// GC = sig(bn(y31))*sig(bn(y32)) over [B,C,W]  (which=1)
__global__ void k_gate_c(float* __restrict__ ws, const float* __restrict__ wc,
                         const float* __restrict__ bc, int which) {
  size_t i = (size_t)blockIdx.x * blockDim.x + threadIdx.x;
  if (i >= BCH) return;
  const int c = (int)((i / 128) % kC);
  const size_t yA = which ? O_Y31 : O_Y21;
  const size_t yB = which ? O_Y32 : O_Y23;
  const int tA = which ? 2 : 0, tB = which ? 3 : 1;
  const float mA = ws[O_ST + 4 + ((size_t)tA * kC + c) * 2 + 0];
  const float rA = ws[O_ST + 4 + ((size_t)tA * kC + c) * 2 + 1];
  const float mB = ws[O_ST + 4 + ((size_t)tB * kC + c) * 2 + 0];
  const float rB = ws[O_ST + 4 + ((size_t)tB * kC + c) * 2 + 1];
  const float wv = wc[c], bv = bc[c];
  float g = sigmoidf_((ws[yA + i] - mA) * rA * wv + bv) *
            sigmoidf_((ws[yB + i] - mB) * rB * wv + bv);
  ws[(which ? O_GC : O_GB) + i] = g;
}

// ---------------------------------------------------------------------------
// Pass E: out = x * (1 + GA[b,h,w]*GB[b,c,h]*GC[b,c,w])   (float4 I/O)
// ---------------------------------------------------------------------------
__global__ void k_final(const float* __restrict__ x, const float* __restrict__ ws,
                        float* __restrict__ out) {
  size_t i4 = (size_t)blockIdx.x * blockDim.x + threadIdx.x;
  size_t base = i4 * 4;
  if (base >= (size_t)kB * kC * kH * kW) return;
  const size_t chw = (size_t)kC * kH * kW, hw = (size_t)kH * kW;
  const size_t b = base / chw;
  size_t rem = base % chw;
  const size_t c = rem / hw;
  rem = rem % hw;
  const size_t h = rem / kW, w0 = rem % kW;

  float4 xv = *(const float4*)(x + base);
  float4 ga = *(const float4*)(ws + O_GA + b * hw + h * kW + w0);
  float gb = ws[O_GB + (b * kC + c) * kH + h];
  float4 gc = *(const float4*)(ws + O_GC + (b * kC + c) * kW + w0);

  float4 o;
  o.x = xv.x * (1.f + ga.x * gb * gc.x);
  o.y = xv.y * (1.f + ga.y * gb * gc.y);
  o.z = xv.z * (1.f + ga.z * gb * gc.z);
  o.w = xv.w * (1.f + ga.w * gb * gc.w);
  *(float4*)(out + base) = o;
}

// ---------------------------------------------------------------------------
// Host entry
// ---------------------------------------------------------------------------
extern "C" void kernel_launch(void* const* d_in, const int* in_sizes, int n_in,
                              void* d_out, int out_size, void* d_ws, size_t ws_size,
                              hipStream_t stream) {
  (void)in_sizes; (void)n_in; (void)out_size; (void)ws_size;
  const float* x = (const float*)d_in[0];
  const float* bn1_w = (const float*)d_in[1];
  const float* bn1_b = (const float*)d_in[2];
  const float* bnc_w = (const float*)d_in[3];
  const float* bnc_b = (const float*)d_in[4];
  float* out = (float*)d_out;
  float* ws = (float*)d_ws;

  k_pool_hw<<<(int)((BHW / 4 + 255) / 256), 256, 0, stream>>>(x, ws + O_XHW);
  k_pool_chw<<<kB * kC, 128, 0, stream>>>(x, ws + O_XCH, ws + O_XCW);
  k_attn<<<kB * 3, 512, 0, stream>>>(ws);
  k_bn_scalar<<<2, 256, 0, stream>>>(ws);
  k_bn_chan<<<4 * kC, 128, 0, stream>>>(ws);
  k_gate_hw<<<(int)((BHW + 255) / 256), 256, 0, stream>>>(ws, bn1_w, bn1_b);
  k_gate_c<<<(int)((BCH + 255) / 256), 256, 0, stream>>>(ws, bnc_w, bnc_b, 0);
  k_gate_c<<<(int)((BCW + 255) / 256), 256, 0, stream>>>(ws, bnc_w, bnc_b, 1);
  k_final<<<(int)(((size_t)kB * kC * kH * kW / 4 + 255) / 256), 256, 0, stream>>>(x, ws, out);
}